// attention3d_3599182594237
// MI455X (gfx1250) — compile-verified
//
#include <hip/hip_runtime.h>
#include <hip/hip_bf16.h>

typedef float v2f __attribute__((ext_vector_type(2)));
typedef float v8f __attribute__((ext_vector_type(8)));

__device__ __forceinline__ v8f wmma_f32x4(v2f a, v2f b, v8f c) {
  // V_WMMA_F32_16X16X4_F32 : D = A(16x4 f32) * B(4x16 f32) + C(16x16 f32)
  return __builtin_amdgcn_wmma_f32_16x16x4_f32(
      /*neg_a=*/false, a, /*neg_b=*/false, b,
      /*c_mod=*/(short)0, c, /*reuse_a=*/false, /*reuse_b=*/false);
}

// ---------------------------------------------------------------------------
// Kernel 1: slice-conv as GEMM.
//   fea[b, axis*32+sl, co] = bias[co] + sum_{ci, i<31, j<31}
//                            w[co,ci,i+1,j+1] * x_axis_slice[ci,i,j]
// Nested (ci, ii, j0) loops: all axis-dependent indexing reduces to three
// uniform shifts/strides hoisted out of the loops; the fully-unrolled inner
// j0 loop has a compile-time-foldable pad mask for 7 of 8 bodies.
// Grid: 240 blocks = (b:2) x (axis:3) x (mt:2) x (nt:20); 8 waves split the
// 128 input channels; partial v8f accumulators reduced through LDS.
// ---------------------------------------------------------------------------
__global__ __launch_bounds__(256) void conv_gemm_kernel(
    const float* __restrict__ x,
    const float* __restrict__ xw, const float* __restrict__ xbias,
    const float* __restrict__ yw, const float* __restrict__ ybias,
    float* __restrict__ fea) {
  const int tile = blockIdx.x;          // 0..239
  const int nt   = tile % 20;
  const int mt   = (tile / 20) & 1;
  const int ba   = tile / 40;           // 0..5
  const int axis = ba % 3;
  const int b    = ba / 3;

  const float* __restrict__ w    = (axis == 0) ? xw : yw;
  const float* __restrict__ bias = (axis == 0) ? xbias : ybias;

  const int lane = threadIdx.x & 31;
  const int wv   = threadIdx.x >> 5;    // wave 0..7
  const int hi   = lane >> 4;           // 0 / 1
  const int hi2  = hi << 1;
  const int l15  = lane & 15;

  const int sl  = mt * 16 + l15;        // A-matrix row  (slice index 0..31)
  const int co  = nt * 16 + l15;        // B-matrix col  (out channel)

  // Axis-uniform addressing:  x_elem(ci,ii,jj) = x[xb0 + ci*32768 +
  //                                                (ii<<sish) + (jj<<sjsh) + sl*ss]
  //   axis0: x[b,ci,sl,ii,jj]  -> sish=5,  sjsh=0, ss=1024
  //   axis1: x[b,ci,ii,sl,jj]  -> sish=10, sjsh=0, ss=32
  //   axis2: x[b,ci,ii,jj,sl]  -> sish=10, sjsh=5, ss=1
  const int sish = (axis == 0) ? 5 : 10;
  const int sjsh = (axis == 2) ? 5 : 0;
  const int ss   = (axis == 0) ? 1024 : ((axis == 1) ? 32 : 1);
  const int xb0  = b * 4194304 + sl * ss;   // + b*128*32768
  const int wco  = co * 131072;

  v8f acc = {};

  const int ci0 = wv * 16;              // 128 input channels / 8 waves
  for (int ci = ci0; ci < ci0 + 16; ++ci) {
    const int abci = xb0 + ci * 32768;
    const int wbci = wco + ci * 1024;
    for (int ii = 0; ii < 31; ++ii) {   // ii==31 is pure padding: skip
      const int ab = abci + (ii << sish);
      const int wb = wbci + ii * 32 + 33;        // w[co,ci,ii+1, jj+1] = w[wb+jj]
      __builtin_prefetch(&w[wb + 32], 0, 1);     // next kernel row (global_prefetch_b8)
#pragma unroll
      for (int j0 = 0; j0 < 32; j0 += 4) {
        v2f af, bf;
#pragma unroll
        for (int e = 0; e < 2; ++e) {
          const int  jj = j0 + hi2 + e;          // this lane's K element
          const bool ok = (jj < 31);             // folds to true for j0 < 28
          const float m = ok ? 1.0f : 0.0f;
          const int  wi = ok ? (wb + jj) : 0;    // clamp: avoid 1-element OOB
          const float av = x[ab + (jj << sjsh)] * m;  // always in-bounds
          const float wvv = w[wi];
          if (e == 0) { af.x = av; bf.x = wvv; }
          else        { af.y = av; bf.y = wvv; }
        }
        acc = wmma_f32x4(af, bf, acc);
      }
    }
  }

  // Reduce the 8 per-wave partial accumulators through LDS.
  __shared__ float red[256 * 8];
#pragma unroll
  for (int r = 0; r < 8; ++r) red[threadIdx.x * 8 + r] = acc[r];
  __syncthreads();

  if (wv == 0) {
#pragma unroll
    for (int r = 0; r < 8; ++r) {
      float s = 0.0f;
      for (int w2 = 0; w2 < 8; ++w2) s += red[(w2 * 32 + lane) * 8 + r];
      const int slo = mt * 16 + r + hi * 8;      // C/D layout: M=r (lo) / r+8 (hi)
      const int coo = nt * 16 + l15;
      fea[(b * 96 + axis * 32 + slo) * 320 + coo] = s + bias[coo];
    }
  }
}

// ---------------------------------------------------------------------------
// Kernel 2/4: generic WMMA GEMM  C[M,N] = A[M,K] @ W[N,K]^T + bias[N]
// One wave per 16x16 tile; K multiple of 4; M,N multiples of 16.
// ---------------------------------------------------------------------------
__global__ __launch_bounds__(32) void gemm_bias_wmma_kernel(
    const float* __restrict__ A, const float* __restrict__ W,
    const float* __restrict__ bias, float* __restrict__ C,
    int M, int N, int K) {
  const int nt    = blockIdx.x;
  const int mtile = blockIdx.y;
  const int lane  = threadIdx.x & 31;
  const int hi    = lane >> 4;
  const int l15   = lane & 15;

  const int row = mtile * 16 + l15;   // A-matrix row for this lane
  const int col = nt * 16 + l15;      // B-matrix col for this lane
  const float* __restrict__ Ar = A + row * K;
  const float* __restrict__ Wr = W + col * K;

  v8f acc = {};
#pragma unroll 4
  for (int k0 = 0; k0 < K; k0 += 4) {
    const int kb = k0 + (hi << 1);
    v2f afrag, bfrag;
    afrag.x = Ar[kb];
    afrag.y = Ar[kb + 1];
    bfrag.x = Wr[kb];
    bfrag.y = Wr[kb + 1];
    acc = wmma_f32x4(afrag, bfrag, acc);
  }

  const float bv = bias[col];
#pragma unroll
  for (int r = 0; r < 8; ++r) {
    const int ro = mtile * 16 + r + hi * 8;
    C[ro * N + col] = acc[r] + bv;
  }
}

// ---------------------------------------------------------------------------
// Kernel 3: fused attention, one block per (b, head). N=96, Dh=64, scale=5^-0.5
// q/k/v and the 96x96 score matrix live in dynamic LDS (110592 B < 320 KB).
// ---------------------------------------------------------------------------
__global__ __launch_bounds__(128) void attn_kernel(
    const float* __restrict__ qkv, float* __restrict__ outb) {
  extern __shared__ float sm[];
  float* qs = sm;            // 96*64
  float* ks = qs + 6144;     // 96*64
  float* vs = ks + 6144;     // 96*64
  float* ps = vs + 6144;     // 96*96
  const int bh  = blockIdx.x;
  const int b   = bh / 5;
  const int h   = bh % 5;
  const int tid = threadIdx.x;

  for (int idx = tid; idx < 6144; idx += 128) {
    const int t = idx >> 6, d = idx & 63;
    const int base = (b * 96 + t) * 960 + h * 64 + d;  // [B,N,3,H,Dh] packing
    qs[idx] = qkv[base];
    ks[idx] = qkv[base + 320];
    vs[idx] = qkv[base + 640];
  }
  __syncthreads();

  if (tid < 96) {
    const float scale = 0.44721359549995793f;  // HEAD_NUM**-0.5 (as in module)
    float mx = -3.402823466e38f;
    for (int m = 0; m < 96; ++m) {
      float s = 0.0f;
#pragma unroll 8
      for (int d = 0; d < 64; ++d) s += qs[tid * 64 + d] * ks[m * 64 + d];
      s *= scale;
      ps[tid * 96 + m] = s;
      mx = fmaxf(mx, s);
    }
    float sum = 0.0f;
    for (int m = 0; m < 96; ++m) {
      const float e = expf(ps[tid * 96 + m] - mx);
      ps[tid * 96 + m] = e;
      sum += e;
    }
    const float inv = 1.0f / sum;
    for (int d = 0; d < 64; ++d) {
      float o = 0.0f;
#pragma unroll 8
      for (int m = 0; m < 96; ++m) o += ps[tid * 96 + m] * vs[m * 64 + d];
      outb[(b * 96 + tid) * 320 + h * 64 + d] = o * inv;  // (0,2,1,3) concat
    }
  }
}

// ---------------------------------------------------------------------------
// Kernel 5: assemble output [B, 128+960, 32,32,32]:
//   c <  128 : copy x
//   c >= 128 : axial broadcast of proj output (+ pos_emb)
// Pure float4 streaming (~285 MB write) -> bandwidth bound.
// ---------------------------------------------------------------------------
__global__ __launch_bounds__(256) void assemble_kernel(
    const float* __restrict__ x, const float* __restrict__ proj,
    const float* __restrict__ pos, float* __restrict__ out) {
  const long gid = (long)blockIdx.x * 256 + threadIdx.x;  // float4 index
  if (gid >= 17825792L) return;                           // 2*1088*32*32*8
  const int z4 = (int)(gid & 7) * 4;
  long t = gid >> 3;
  const int Y = (int)(t & 31); t >>= 5;
  const int X = (int)(t & 31); t >>= 5;
  const int ti = (int)t;
  const int c = ti % 1088;
  const int b = ti / 1088;

  float4 r;
  if (c < 128) {
    r = *(const float4*)(x + ((((b * 128 + c) * 32 + X) * 32 + Y) * 32 + z4));
  } else {
    const int cc  = c - 128;
    const int grp = cc / 320;
    const int ch  = cc - grp * 320;
    const float4 pe = *(const float4*)(pos + ((X * 32 + Y) * 32 + z4));
    if (grp == 0) {
      const float s = proj[(b * 96 + X) * 320 + ch];
      r.x = s + pe.x; r.y = s + pe.y; r.z = s + pe.z; r.w = s + pe.w;
    } else if (grp == 1) {
      const float s = proj[(b * 96 + 32 + Y) * 320 + ch];
      r.x = s + pe.x; r.y = s + pe.y; r.z = s + pe.z; r.w = s + pe.w;
    } else {
      const int base = (b * 96 + 64 + z4) * 320 + ch;
      r.x = proj[base]       + pe.x;
      r.y = proj[base + 320] + pe.y;
      r.z = proj[base + 640] + pe.z;
      r.w = proj[base + 960] + pe.w;
    }
  }
  *(float4*)(out + gid * 4) = r;
}

// ---------------------------------------------------------------------------
extern "C" void kernel_launch(void* const* d_in, const int* in_sizes, int n_in,
                              void* d_out, int out_size, void* d_ws, size_t ws_size,
                              hipStream_t stream) {
  (void)in_sizes; (void)n_in; (void)out_size; (void)ws_size;

  const float* x       = (const float*)d_in[0];
  const float* xconv_w = (const float*)d_in[1];
  const float* xconv_b = (const float*)d_in[2];
  const float* yconv_w = (const float*)d_in[3];
  const float* yconv_b = (const float*)d_in[4];
  const float* qkv_w   = (const float*)d_in[5];
  const float* qkv_b   = (const float*)d_in[6];
  const float* proj_w  = (const float*)d_in[7];
  const float* proj_b  = (const float*)d_in[8];
  const float* pos_emb = (const float*)d_in[9];
  float* out = (float*)d_out;

  // Workspace layout (floats): fea[192*320] | qkv[192*960] | attn[192*320] | proj[192*320]
  float* ws      = (float*)d_ws;
  float* fea     = ws;
  float* qkvbuf  = ws + 61440;
  float* attnout = ws + 245760;
  float* projout = ws + 307200;   // total 368640 floats = 1.44 MB

  // 1) slice-convs as WMMA GEMM -> fea [B, 96, 320]
  conv_gemm_kernel<<<240, 256, 0, stream>>>(x, xconv_w, xconv_b,
                                            yconv_w, yconv_b, fea);
  // 2) qkv = fea @ qkv_w^T + qkv_b   (M=192, N=960, K=320)
  gemm_bias_wmma_kernel<<<dim3(60, 12), 32, 0, stream>>>(
      fea, qkv_w, qkv_b, qkvbuf, 192, 960, 320);
  // 3) multi-head attention (10 blocks = B*H)
  attn_kernel<<<10, 128, 110592, stream>>>(qkvbuf, attnout);
  // 4) proj = attn @ proj_w^T + proj_b (M=192, N=320, K=320)
  gemm_bias_wmma_kernel<<<dim3(20, 12), 32, 0, stream>>>(
      attnout, proj_w, proj_b, projout, 192, 320, 320);
  // 5) assemble [2, 1088, 32,32,32]
  assemble_kernel<<<69632, 256, 0, stream>>>(x, projout, pos_emb, out);
}